// fmmLayerWindow_59906203845037
// MI455X (gfx1250) — compile-verified
//
#include <hip/hip_runtime.h>
#include <math.h>

typedef __attribute__((ext_vector_type(2))) float v2f;
typedef __attribute__((ext_vector_type(8))) float v8f;

#define NTOT 512   // NCELLS * NP = 64 * 8

// Branchless erf: Abramowitz & Stegun 7.1.26, |error| < 1.5e-7.
// Uses one v_rcp, one raw v_exp (exp2 underflows cleanly for large |x|),
// six FMAs and a sign select -> no EXEC manipulation in the hot loop.
__device__ __forceinline__ float erf_fast(float z) {
  float az = fabsf(z);
  float t  = __builtin_amdgcn_rcpf(fmaf(0.3275911f, az, 1.0f));
  float p  = fmaf(t, 1.061405429f, -1.453152027f);
  p = fmaf(t, p, 1.421413741f);
  p = fmaf(t, p, -0.284496736f);
  p = fmaf(t, p, 0.254829592f);
  float e = __builtin_amdgcn_exp2f(-z * z * 1.442695041f);
  float ea = fmaf(-(p * t), e, 1.0f);        // erf(|z|)
  return (z < 0.0f) ? -ea : ea;              // v_cndmask, not a branch
}

// Four kernel bases for one pair distance d (diag => contribute 0).
// window = 1 - 0.5*erfc(d/0.5 - 3) = 0.5 + 0.5*erf(2d - 3)
// b0 = w/d, b1 = w/sqrt(d), b2 = w/d^2, b3 = w*i0e(d)
__device__ __forceinline__ void pair_bases(float d, bool diag,
                                           float& b0, float& b1,
                                           float& b2, float& b3) {
  float dd  = diag ? 1.0f : d;                 // Dsafe (cndmask)
  float inv = __builtin_amdgcn_rcpf(dd);
  float rs  = __builtin_amdgcn_rsqf(dd);
  float w   = fmaf(0.5f, erf_fast(fmaf(2.0f, dd, -3.0f)), 0.5f);
  w = diag ? 0.0f : w;                         // mask diagonal (cndmask)

  // i0e(dd): Abramowitz & Stegun 9.8.1 / 9.8.2, branchless select
  float t = dd * 0.2666666667f;                // dd/3.75
  t = t * t;
  float ps = fmaf(t, 0.0045813f, 0.0360768f);
  ps = fmaf(t, ps, 0.2659732f);
  ps = fmaf(t, ps, 1.2067492f);
  ps = fmaf(t, ps, 3.0899424f);
  ps = fmaf(t, ps, 3.5156229f);
  ps = fmaf(t, ps, 1.0f);
  float i0_small = ps * __builtin_amdgcn_exp2f(dd * -1.442695041f);

  float y = 3.75f * inv;                       // 3.75/dd
  float pl = fmaf(y, 0.00392377f, -0.01647633f);
  pl = fmaf(y, pl, 0.02635537f);
  pl = fmaf(y, pl, -0.02057706f);
  pl = fmaf(y, pl, 0.00916281f);
  pl = fmaf(y, pl, -0.00157565f);
  pl = fmaf(y, pl, 0.00225319f);
  pl = fmaf(y, pl, 0.01328592f);
  pl = fmaf(y, pl, 0.39894228f);
  float i0_large = pl * rs;

  float i0e = (dd < 3.75f) ? i0_small : i0_large;   // cndmask

  b0 = w * inv;
  b1 = w * rs;
  b2 = (w * inv) * inv;
  b3 = w * i0e;
}

__global__ __launch_bounds__(256)
void fmmLayerWindow_kernel(const float* __restrict__ R,      // (Ns, 512)
                           const float* __restrict__ stdv,   // (4,)
                           const float* __restrict__ bias,   // (4,)
                           float* __restrict__ out) {        // (Ns, 512, 4)
  __shared__ float Rsh[NTOT];

  const int samp = blockIdx.y;
  const int t    = threadIdx.x;

  // Stage this sample's 512 coordinates into LDS (2 KB).
  Rsh[t]       = R[samp * NTOT + t];
  Rsh[t + 256] = R[samp * NTOT + 256 + t];
  __syncthreads();

  const int lane = t & 31;        // wave32
  const int wave = t >> 5;        // 8 waves / block
  const int half = lane >> 4;     // 0 -> supplies K=0,1 ; 1 -> supplies K=2,3
  const int m    = lane & 15;     // A-matrix row M
  const int iIdx = blockIdx.x * 128 + wave * 16 + m;   // global i for this lane
  const float ri = Rsh[iIdx];

  v8f acc0 = {}, acc1 = {}, acc2 = {}, acc3 = {};
  v2f ones; ones.x = 1.0f; ones.y = 1.0f;              // B = 4x16 ones (layout-free)

  const float2* R2 = reinterpret_cast<const float2*>(Rsh);

  // j loop in chunks of 4 (K dimension of V_WMMA_F32_16X16X4_F32).
  // Body is a single straight-line block: EXEC stays all-1s at every WMMA.
  for (int jc = 0; jc < NTOT; jc += 4) {
    float2 rj = R2[(jc >> 1) + half];    // Rsh[jc+2*half], Rsh[jc+2*half+1]
    const int jA = jc + 2 * half;

    float d0 = fabsf(rj.x - ri);
    float d1 = fabsf(rj.y - ri);

    float a0x, a1x, a2x, a3x, a0y, a1y, a2y, a3y;
    pair_bases(d0, iIdx == jA,     a0x, a1x, a2x, a3x);
    pair_bases(d1, iIdx == jA + 1, a0y, a1y, a2y, a3y);

    v2f A;
    // D[m,n] += sum_k A[m,k] * 1 : matrix pipe does the j-reduction,
    // co-executing with the VALU transcendental stream.
    A.x = a0x; A.y = a0y;
    acc0 = __builtin_amdgcn_wmma_f32_16x16x4_f32(false, A, false, ones,
                                                 (short)0, acc0, false, false);
    A.x = a1x; A.y = a1y;
    acc1 = __builtin_amdgcn_wmma_f32_16x16x4_f32(false, A, false, ones,
                                                 (short)0, acc1, false, false);
    A.x = a2x; A.y = a2y;
    acc2 = __builtin_amdgcn_wmma_f32_16x16x4_f32(false, A, false, ones,
                                                 (short)0, acc2, false, false);
    A.x = a3x; A.y = a3y;
    acc3 = __builtin_amdgcn_wmma_f32_16x16x4_f32(false, A, false, ones,
                                                 (short)0, acc3, false, false);
  }

  // Epilogue: out = |std_k| * (sum - (N-1)*bias_k).
  const float s0 = fabsf(stdv[0]), s1 = fabsf(stdv[1]);
  const float s2 = fabsf(stdv[2]), s3 = fabsf(stdv[3]);
  const float c0 = s0 * 511.0f * bias[0];
  const float c1 = s1 * 511.0f * bias[1];
  const float c2 = s2 * 511.0f * bias[2];
  const float c3 = s3 * 511.0f * bias[3];

  // C/D layout: VGPR r holds D[M=r, N=lane] (lanes 0-15) and D[M=8+r, ...]
  // (lanes 16-31); all 16 columns are identical, so lanes 0 and 16 own
  // rows 0..7 and 8..15 of this wave's tile respectively.
  if (m == 0) {
    const int rowBase = blockIdx.x * 128 + wave * 16 + 8 * half;
    float4* out4 = reinterpret_cast<float4*>(out);
#pragma unroll
    for (int r = 0; r < 8; ++r) {
      const int i = rowBase + r;
      float4 v;
      v.x = fmaf(s0, acc0[r], -c0);
      v.y = fmaf(s1, acc1[r], -c1);
      v.z = fmaf(s2, acc2[r], -c2);
      v.w = fmaf(s3, acc3[r], -c3);
      out4[samp * NTOT + i] = v;
    }
  }
}

extern "C" void kernel_launch(void* const* d_in, const int* in_sizes, int n_in,
                              void* d_out, int out_size, void* d_ws, size_t ws_size,
                              hipStream_t stream) {
  const float* R    = (const float*)d_in[0];
  const float* stdv = (const float*)d_in[1];
  const float* bias = (const float*)d_in[2];
  float* out        = (float*)d_out;

  const int nsamples = in_sizes[0] / NTOT;     // 128
  dim3 grid(4, nsamples);                      // 4 i-tiles of 128 rows x samples
  fmmLayerWindow_kernel<<<grid, 256, 0, stream>>>(R, stdv, bias, out);
}